// dtsemnet_topk_actor_14216341750428
// MI455X (gfx1250) — compile-verified
//
#include <hip/hip_runtime.h>
#include <hip/hip_bf16.h>

// ---------------------------------------------------------------------------
// DT-SemNet top-k actor forward, MI455X (gfx1250, wave32, WMMA + TDM).
// Forward value reduces to: pick argmax leaf, apply that leaf's linear head.
//   stage 0 (prep): fold path matrix:  WnP = Wn @ P  [128,16], bnP = bn @ P [16]
//   x tile staged into LDS by the Tensor Data Mover (one 16x128 2D tile per
//   wave, D# padding gives the 132-float LDS row stride for bank spread)
//   stage 1: leaf_logits = x @ WnP + bnP via V_WMMA_F32_16X16X4_F32, argmax
//   stage 2: compact rows by leaf, dense WMMA per (leaf, 16-row tile)
// ---------------------------------------------------------------------------

typedef __attribute__((ext_vector_type(2))) float v2f;
typedef __attribute__((ext_vector_type(8))) float v8f;
typedef unsigned int u32;
typedef __attribute__((ext_vector_type(4))) u32 v4u;
typedef __attribute__((ext_vector_type(8))) u32 v8u;

#define DDIM      128
#define LLEAF     16
#define ROWS      128          // rows per workgroup
#define XS_STRIDE 132          // padded x row stride in LDS (floats) = 128 + pad 4
#define WP_STRIDE 130          // padded WnP^T row stride (floats)

#define LDS_FLOATS (ROWS*XS_STRIDE + LLEAF*WP_STRIDE + LLEAF)
#define LDS_INTS   (ROWS + ROWS + LLEAF + LLEAF + 32 + 32 + 1 + ROWS)  // 481
#define LDS_BYTES  ((LDS_FLOATS + LDS_INTS) * 4)

// ---- prep: fold ±path matrix into node weights ----------------------------
extern "C" __global__ void dts_prep(const float* __restrict__ Wn,
                                    const float* __restrict__ bn,
                                    float* __restrict__ ws) {
  int t = threadIdx.x;  // 256 threads
  #pragma unroll
  for (int j = 0; j < 8; ++j) {
    int e = t * 8 + j;              // e = d*16 + leaf
    int d = e >> 4, leaf = e & 15;
    float s = 0.f; int node = 0;
    #pragma unroll
    for (int lvl = 0; lvl < 4; ++lvl) {
      int bit = (leaf >> (3 - lvl)) & 1;
      s += (bit ? -1.f : 1.f) * Wn[d * 15 + node];
      node = 2 * node + 1 + bit;
    }
    ws[e] = s;
  }
  if (t < LLEAF) {
    float s = 0.f; int node = 0;
    #pragma unroll
    for (int lvl = 0; lvl < 4; ++lvl) {
      int bit = (t >> (3 - lvl)) & 1;
      s += (bit ? -1.f : 1.f) * bn[node];
      node = 2 * node + 1 + bit;
    }
    ws[DDIM * LLEAF + t] = s;
  }
}

// ---- main fused kernel ----------------------------------------------------
extern "C" __global__ void __launch_bounds__(256)
dts_main(const float* __restrict__ x, const float* __restrict__ Wl,
         const float* __restrict__ bl, const float* __restrict__ wnp,
         float* __restrict__ out, int Brows) {
  extern __shared__ float lds[];
  float* xs   = lds;                                // [ROWS][XS_STRIDE]
  float* wnpT = lds + ROWS * XS_STRIDE;             // [16][WP_STRIDE]
  float* bnp  = wnpT + LLEAF * WP_STRIDE;           // [16]
  int* ib     = (int*)(bnp + LLEAF);
  int* lidx   = ib;                                 // [128] chosen leaf per row
  int* lpos   = ib + ROWS;                          // [128] slot within leaf
  int* cnt    = ib + 2 * ROWS;                      // [16]
  int* off    = cnt + LLEAF;                        // [16]
  int* tleaf  = off + LLEAF;                        // [32] tile worklist
  int* ttb    = tleaf + 32;                         // [32]
  int* ntp    = ttb + 32;                           // [1]
  int* perm   = ntp + 1;                            // [128]

  const int tid  = threadIdx.x;
  const int lane = tid & 31;
  const int wave = tid >> 5;          // 8 waves
  const int h    = lane >> 4;         // lane half (K pairing)
  const int n    = lane & 15;         // column / row-in-tile index
  const int rowBase = blockIdx.x * ROWS;

  if (tid < LLEAF) cnt[tid] = 0;

  // --- TDM: each wave DMAs its 16x128 f32 x-tile into LDS ----------------
  // D# group0: count=1 | lds_addr | global_addr(57b) | type=2
  // D# group1: data_size=4B, pad_enable, pad_interval=128 dw (code 6),
  //            pad_amount=4 dw (code 3)  -> LDS row stride 132 floats.
  {
    unsigned long long gaddr =
        (unsigned long long)(x + (size_t)(rowBase + wave * 16) * DDIM);
    v4u g0 = { 1u,                                        // count = 1
               (u32)(wave * 16 * XS_STRIDE * 4),          // lds byte address
               (u32)(gaddr & 0xffffffffu),                // global_addr[31:0]
               (u32)((gaddr >> 32) & 0x01ffffffu) | (2u << 30) }; // [56:32]|type
    v8u g1 = { (2u << 16) | (1u << 20) | (6u << 22) | (3u << 25),
               (u32)(DDIM) << 16,   // tensor_dim0[15:0] @ bits 63:48
               16u << 16,           // tensor_dim1[15:0] @ bits 95:80 (16 rows)
               (u32)(DDIM) << 16,   // tile_dim0 @ bits 127:112
               16u,                 // tile_dim1 @ bits 143:128
               (u32)(DDIM),         // tensor_dim0_stride[31:0] @ bits 191:160
               0u, 0u };
    asm volatile("tensor_load_to_lds %0, %1"
                 :: "s"(g0), "s"(g1)
                 : "memory");
  }

  // --- cooperative loads: folded node weights (small, regular path) -------
  #pragma unroll
  for (int j = 0; j < 8; ++j) {               // 2048 entries of WnP^T
    int e = tid + 256 * j;
    int leaf = e >> 7, d = e & 127;
    wnpT[leaf * WP_STRIDE + d] = wnp[d * LLEAF + leaf];
  }
  if (tid < LLEAF) bnp[tid] = wnp[DDIM * LLEAF + tid];
  __builtin_prefetch(Wl + tid * 128, 0, 0);   // pull 128KB Wl toward caches

  __builtin_amdgcn_s_wait_tensorcnt(0);       // my tile is in LDS
  __syncthreads();                            // everyone's tile is in LDS

  // --- stage 1: leaf logits via WMMA f32 16x16x4, argmax per row ----------
  {
    v8f acc = {0.f, 0.f, 0.f, 0.f, 0.f, 0.f, 0.f, 0.f};
    const float* xrow = xs + (wave * 16 + n) * XS_STRIDE;  // A: M = lane&15
    const float* wcol = wnpT + n * WP_STRIDE;              // B: N = lane&15
    #pragma unroll 8
    for (int ks = 0; ks < 32; ++ks) {
      int kb = ks * 4 + 2 * h;                 // K = kb + vgpr (ISA A layout)
      v2f a = *(const v2f*)(xrow + kb);
      v2f b = *(const v2f*)(wcol + kb);
      acc = __builtin_amdgcn_wmma_f32_16x16x4_f32(
          false, a, false, b, (short)0, acc, false, false);
    }
    float bb = bnp[n];
    #pragma unroll
    for (int v = 0; v < 8; ++v) {              // row m = v + 8*h of this tile
      float val = acc[v] + bb;
      int bi = n;
      #pragma unroll
      for (int m = 1; m < 16; m <<= 1) {       // argmax over 16 lanes/half
        float ov = __shfl_xor(val, m, 32);
        int   oi = __shfl_xor(bi,  m, 32);
        if (ov > val || (ov == val && oi < bi)) { val = ov; bi = oi; }
      }
      if (n == 0) {
        int r = wave * 16 + v + 8 * h;
        lidx[r] = bi;
        lpos[r] = atomicAdd(&cnt[bi], 1);
      }
    }
  }
  __syncthreads();

  // --- prefix sums + tile worklist ----------------------------------------
  if (tid == 0) {
    int o = 0, t = 0;
    for (int l = 0; l < LLEAF; ++l) {
      off[l] = o;
      int c = cnt[l];
      o += c;
      for (int tb = 0; tb * 16 < c; ++tb) { tleaf[t] = l; ttb[t] = tb; ++t; }
    }
    ntp[0] = t;
  }
  __syncthreads();
  if (tid < ROWS) perm[off[lidx[tid]] + lpos[tid]] = tid;
  __syncthreads();

  // --- stage 2: selected-leaf heads, WMMA over compacted row tiles --------
  const int ntiles = ntp[0];
  for (int t = wave; t < ntiles; t += 8) {
    const int leaf = tleaf[t], tb = ttb[t];
    const int o0 = off[leaf], c = cnt[leaf];
    int mr = tb * 16 + n;
    int lr = perm[o0 + (mr < c ? mr : 0)];     // dup row 0 on ragged tail
    const float* xrow = xs + lr * XS_STRIDE;
    const float* wlp  = Wl + leaf * (DDIM * LLEAF);
    v8f acc = {0.f, 0.f, 0.f, 0.f, 0.f, 0.f, 0.f, 0.f};
    #pragma unroll 4
    for (int ks = 0; ks < 32; ++ks) {
      int kb = ks * 4 + 2 * h;
      v2f a = *(const v2f*)(xrow + kb);
      v2f b;
      b.x = wlp[kb * LLEAF + n];               // B[K=kb  ][N=n]
      b.y = wlp[(kb + 1) * LLEAF + n];         // B[K=kb+1][N=n]
      acc = __builtin_amdgcn_wmma_f32_16x16x4_f32(
          false, a, false, b, (short)0, acc, false, false);
    }
    float bias = bl[leaf * LLEAF + n];
    #pragma unroll
    for (int v = 0; v < 8; ++v) {
      int mrow = tb * 16 + v + 8 * h;
      if (mrow < c) {
        int grow = rowBase + perm[o0 + mrow];
        float val = acc[v] + bias;
        if (n < 8) {
          out[(size_t)grow * 8 + n] = val;                      // mean
        } else {
          float th = tanhf(val);                                // log_std
          out[(size_t)Brows * 8 + (size_t)grow * 8 + (n - 8)] =
              -5.0f + 3.5f * (th + 1.0f);
        }
      }
    }
  }
}

// ---------------------------------------------------------------------------
extern "C" void kernel_launch(void* const* d_in, const int* in_sizes, int n_in,
                              void* d_out, int out_size, void* d_ws, size_t ws_size,
                              hipStream_t stream) {
  const float* x  = (const float*)d_in[0];
  const float* Wn = (const float*)d_in[1];
  const float* bn = (const float*)d_in[2];
  const float* Wl = (const float*)d_in[3];
  const float* bl = (const float*)d_in[4];
  float* out = (float*)d_out;
  float* ws  = (float*)d_ws;     // WnP [128*16] + bnP [16]

  const int B = in_sizes[0] / DDIM;

  dts_prep<<<1, 256, 0, stream>>>(Wn, bn, ws);

  (void)hipFuncSetAttribute((const void*)dts_main,
                            hipFuncAttributeMaxDynamicSharedMemorySize,
                            LDS_BYTES);
  dts_main<<<B / ROWS, 256, LDS_BYTES, stream>>>(x, Wl, bl, ws, out, B);
}